// LinearAttention2d_19232863551967
// MI455X (gfx1250) — compile-verified
//
#include <hip/hip_runtime.h>

typedef float v2f __attribute__((ext_vector_type(2)));
typedef float v8f __attribute__((ext_vector_type(8)));
typedef unsigned int u32x4 __attribute__((ext_vector_type(4)));
typedef unsigned int u32x8 __attribute__((ext_vector_type(8)));

#define NB       4
#define SEQ      4096
#define NHEADS   16
#define DDIM     64
#define MDIM     64
#define NHTOT    (NB * NHEADS)        // 64
#define NSPLIT   8
#define S_SPLIT  (SEQ / NSPLIT)       // 512
#define CHUNK    64
#define KV_ELEMS (MDIM * DDIM)        // 4096
#define PART_STRIDE (KV_ELEMS + DDIM) // 4160 floats: KV[m][d] then Ksum[d]
#define LBLK     128
#define EPSV     1e-6f

// ---- TDM: DMA one 64x64 f32 tile (row stride in elements) into LDS with
//      per-row padding. D# per CDNA5 ISA ch.8: group0 = {count, lds_addr,
//      global_addr, type=2}; group1 = {data_size=4B, pad ctl, dims, strides}.
//      pad_interval code 5 = 64 dwords; pad_amount code: 1 = 2 dw, 3 = 4 dw.
__device__ __forceinline__ void tdm_load_tile64(unsigned int lds_addr,
                                                const void* gptr,
                                                unsigned int row_stride_elems,
                                                unsigned int pad_amount_code) {
    const unsigned long long ga = (unsigned long long)(uintptr_t)gptr;
    u32x4 g0;
    g0[0] = 1u;                                               // count=1 (valid D#)
    g0[1] = lds_addr;                                         // LDS byte address
    g0[2] = (unsigned int)ga;                                 // global_addr[31:0]
    g0[3] = (unsigned int)((ga >> 32) & 0x01FFFFFFu)          // global_addr[56:32]
          | (2u << 30);                                       // type=2 (image)
    u32x8 g1;
    g1[0] = (2u << 16)                                        // data_size = 4B
          | (1u << 20)                                        // pad_enable
          | (5u << 22)                                        // pad_interval = 64 dw
          | (pad_amount_code << 25);                          // pad_amount
    g1[1] = 64u << 16;                                        // tensor_dim0 = 64 (lo16)
    g1[2] = 64u << 16;                                        // tensor_dim1 = 64 (lo16)
    g1[3] = 64u << 16;                                        // tile_dim0 = 64
    g1[4] = 64u;                                              // tile_dim1 = 64
    g1[5] = row_stride_elems;                                 // dim0_stride (lo32)
    g1[6] = 0u;
    g1[7] = 0u;
    asm volatile("tensor_load_to_lds %0, %1" :: "s"(g0), "s"(g1) : "memory");
}

// -------- Phase 1: KV = V^T x K and Ksum = sum_s K, split over s ----------
__global__ __launch_bounds__(256) void la_phase1(const float* __restrict__ Kg,
                                                 const float* __restrict__ Vg,
                                                 float* __restrict__ part) {
    __shared__ float Kl[2][CHUNK][DDIM + 4];   // TDM pads rows 64dw + 4dw
    __shared__ float Vl[2][CHUNK][MDIM + 4];
    __shared__ float kred[4][DDIM];

    const int nh    = blockIdx.x / NSPLIT;
    const int split = blockIdx.x % NSPLIT;
    const int b  = nh / NHEADS;
    const int hh = nh % NHEADS;

    const int tid  = threadIdx.x;
    const int wid  = tid >> 5;
    const int lane = tid & 31;
    const int half = lane >> 4;
    const int lm   = lane & 15;

    const int mt  = wid >> 1;          // wave's m-tile (A fragment shared)
    const int dt0 = (wid & 1) * 2;     // wave's two d-tiles

    const size_t rowStride = (size_t)NHEADS * DDIM;     // 1024 floats per s step
    const float* kbase = Kg + ((size_t)b * SEQ * NHEADS + hh) * DDIM;
    const float* vbase = Vg + ((size_t)b * SEQ * NHEADS + hh) * MDIM;

    const unsigned int kaddr[2] = { (unsigned int)(uintptr_t)&Kl[0][0][0],
                                    (unsigned int)(uintptr_t)&Kl[1][0][0] };
    const unsigned int vaddr[2] = { (unsigned int)(uintptr_t)&Vl[0][0][0],
                                    (unsigned int)(uintptr_t)&Vl[1][0][0] };

    v8f acc0 = {};
    v8f acc1 = {};
    float ksum = 0.0f;
    const int sumd = tid & 63;
    const int sumr = tid >> 6;

    const int s_begin = split * S_SPLIT;
    const int nchunks = S_SPLIT / CHUNK;                // 8

    // prologue: DMA chunk 0 into buffer 0 (wave 0 drives the TDM)
    if (wid == 0) {
        tdm_load_tile64(kaddr[0], kbase + (size_t)s_begin * rowStride, 1024u, 3u);
        tdm_load_tile64(vaddr[0], vbase + (size_t)s_begin * rowStride, 1024u, 3u);
    }

    for (int c = 0; c < nchunks; ++c) {
        const int cur = c & 1;
        if (wid == 0) {
            if (c + 1 < nchunks) {
                const size_t snext = (size_t)(s_begin + (c + 1) * CHUNK) * rowStride;
                tdm_load_tile64(kaddr[cur ^ 1], kbase + snext, 1024u, 3u);
                tdm_load_tile64(vaddr[cur ^ 1], vbase + snext, 1024u, 3u);
                __builtin_amdgcn_s_wait_tensorcnt((short)2);   // chunk c landed
            } else {
                __builtin_amdgcn_s_wait_tensorcnt((short)0);
            }
        }
        __syncthreads();                                 // publish chunk c to WG

        #pragma unroll
        for (int ks = 0; ks < CHUNK; ks += 4) {
            const int sr = ks + 2 * half;
            v2f a;                                       // A = V^T tile (m x s)
            a.x = Vl[cur][sr    ][mt * 16 + lm];
            a.y = Vl[cur][sr + 1][mt * 16 + lm];
            v2f b0;                                      // B = K tile (s x d)
            b0.x = Kl[cur][sr    ][dt0 * 16 + lm];
            b0.y = Kl[cur][sr + 1][dt0 * 16 + lm];
            v2f b1;
            b1.x = Kl[cur][sr    ][(dt0 + 1) * 16 + lm];
            b1.y = Kl[cur][sr + 1][(dt0 + 1) * 16 + lm];
            acc0 = __builtin_amdgcn_wmma_f32_16x16x4_f32(false, a, false, b0,
                                                         (short)0, acc0, false, false);
            acc1 = __builtin_amdgcn_wmma_f32_16x16x4_f32(false, a, false, b1,
                                                         (short)0, acc1, false, false);
        }
        #pragma unroll
        for (int r = sumr; r < CHUNK; r += 4) ksum += Kl[cur][r][sumd];
        __syncthreads();                                 // done reading buffer cur
    }

    kred[sumr][sumd] = ksum;
    __syncthreads();

    float* pbase = part + (size_t)blockIdx.x * PART_STRIDE;
    #pragma unroll
    for (int v = 0; v < 8; ++v) {
        const int m0 = mt * 16 + v + 8 * half;
        pbase[m0 * DDIM + dt0 * 16 + lm]       = acc0[v];
        pbase[m0 * DDIM + (dt0 + 1) * 16 + lm] = acc1[v];
    }
    if (tid < DDIM)
        pbase[KV_ELEMS + tid] = kred[0][tid] + kred[1][tid] + kred[2][tid] + kred[3][tid];
}

// -------- Reduce the NSPLIT partials (deterministic, no float atomics) ----
__global__ __launch_bounds__(256) void la_reduce(const float* __restrict__ part,
                                                 float* __restrict__ fin) {
    const int nh = blockIdx.x;
    for (int idx = threadIdx.x; idx < PART_STRIDE; idx += 256) {
        float s = 0.0f;
        #pragma unroll
        for (int sp = 0; sp < NSPLIT; ++sp)
            s += part[((size_t)nh * NSPLIT + sp) * PART_STRIDE + idx];
        fin[(size_t)nh * PART_STRIDE + idx] = s;
    }
}

// -------- Phase 2: out = Z * (Q x KV^T) ----------------------------------
__global__ __launch_bounds__(256) void la_phase2(const float* __restrict__ Qg,
                                                 const float* __restrict__ fin,
                                                 float* __restrict__ Og) {
    __shared__ float KVl[MDIM][DDIM + 2];   // TDM pad 2dw -> conflict-free b64
    __shared__ float Ksl[DDIM];
    __shared__ float Zl[LBLK];

    const int nLB = SEQ / LBLK;             // 32
    const int nh  = blockIdx.x / nLB;
    const int lb  = blockIdx.x % nLB;
    const int b   = nh / NHEADS;
    const int hh  = nh % NHEADS;
    const int lbase = lb * LBLK;

    const int tid  = threadIdx.x;
    const int wid  = tid >> 5;
    const int lane = tid & 31;
    const int half = lane >> 4;
    const int lm   = lane & 15;

    const float* fbase = fin + (size_t)nh * PART_STRIDE;

    // one TDM DMA stages the whole 16KB KV tile (contiguous rows, stride 64)
    if (wid == 0) {
        tdm_load_tile64((unsigned int)(uintptr_t)&KVl[0][0], fbase, 64u, 1u);
        __builtin_amdgcn_s_wait_tensorcnt((short)0);
    }
    if (tid < DDIM) Ksl[tid] = fbase[KV_ELEMS + tid];
    __syncthreads();

    const size_t rowStride = (size_t)NHEADS * DDIM;
    const float* qbase = Qg + ((size_t)b * SEQ * NHEADS + hh) * DDIM;

    // normalizer Z[l] = 1 / (Q[l].Ksum + eps)
    if (tid < LBLK) {
        const float* qrow = qbase + (size_t)(lbase + tid) * rowStride;
        float dot = 0.0f;
        #pragma unroll
        for (int d = 0; d < DDIM; ++d) dot += qrow[d] * Ksl[d];
        Zl[tid] = 1.0f / (dot + EPSV);
    }
    __syncthreads();

    // wave w owns l-strip [lbase + w*16, +16), full 64-wide m
    const int lrow = lbase + wid * 16 + lm;
    const float* qlane = qbase + (size_t)lrow * rowStride;

    v8f acc[4] = {};
    #pragma unroll
    for (int d0 = 0; d0 < DDIM; d0 += 4) {
        const v2f a = *(const v2f*)(qlane + d0 + 2 * half);   // A = Q fragment
        #pragma unroll
        for (int t = 0; t < 4; ++t) {
            v2f bf;                                           // B = KV^T fragment
            bf.x = KVl[t * 16 + lm][d0 + 2 * half];
            bf.y = KVl[t * 16 + lm][d0 + 2 * half + 1];
            acc[t] = __builtin_amdgcn_wmma_f32_16x16x4_f32(false, a, false, bf,
                                                           (short)0, acc[t], false, false);
        }
    }

    float* obase = Og + ((size_t)b * SEQ * NHEADS + hh) * MDIM;
    #pragma unroll
    for (int t = 0; t < 4; ++t) {
        #pragma unroll
        for (int v = 0; v < 8; ++v) {
            const int l = wid * 16 + v + 8 * half;            // l within block
            const float z = Zl[l];
            obase[(size_t)(lbase + l) * rowStride + t * 16 + lm] = acc[t][v] * z;
        }
    }
}

extern "C" void kernel_launch(void* const* d_in, const int* in_sizes, int n_in,
                              void* d_out, int out_size, void* d_ws, size_t ws_size,
                              hipStream_t stream) {
    const float* Q = (const float*)d_in[0];
    const float* K = (const float*)d_in[1];
    const float* V = (const float*)d_in[2];
    float* out  = (float*)d_out;
    float* part = (float*)d_ws;                                   // 8.5 MB partials
    float* fin  = part + (size_t)NHTOT * NSPLIT * PART_STRIDE;    // + 1.06 MB final

    la_phase1<<<NHTOT * NSPLIT, 256, 0, stream>>>(K, V, part);
    la_reduce<<<NHTOT, 256, 0, stream>>>(part, fin);
    la_phase2<<<NHTOT * (SEQ / LBLK), 256, 0, stream>>>(Q, fin, out);
}